// RPN_35613868819177
// MI455X (gfx1250) — compile-verified
//
#include <hip/hip_runtime.h>
#include <hip/hip_bf16.h>
#include <math.h>

// ---------------------------------------------------------------------------
// RPN pipeline for MI455X (gfx1250, wave32).
// conv3x3 (implicit GEMM, V_WMMA_F32_16X16X4_F32, tap-major K) -> heads ->
// decode -> NMS.
// ---------------------------------------------------------------------------

typedef float v2f __attribute__((ext_vector_type(2)));
typedef float v8f __attribute__((ext_vector_type(8)));

#define RPN_B     4
#define RPN_C     512
#define RPN_HW    625          // 25*25
#define RPN_M     2500         // B*H*W
#define RPN_N     512
#define RPN_K     4608         // 512*9
#define RPN_KB    32           // K block staged in LDS
#define RPN_NBOX  5625         // 625*9 proposals per batch
#define RPN_MW    176          // ceil(5625/32) mask words per row
#define NMS_THR   0.7f

// ---------------------------------------------------------------------------
// 0. Repack conv weights: conv_w[o][c][ky][kx] -> wt2[o][tap*512 + c]
//    (tap-major K so the GEMM K loop needs no div/mod and B stages as
//     contiguous float4 in k).
// ---------------------------------------------------------------------------
__global__ void wtrans_kernel(const float* __restrict__ conv_w,
                              float* __restrict__ wt2) {
  int idx = blockIdx.x * blockDim.x + threadIdx.x;
  if (idx >= RPN_N * RPN_K) return;
  int o   = idx / RPN_K;
  int r   = idx - o * RPN_K;     // tap*512 + c
  int tap = r >> 9;              // 0..8
  int c   = r & 511;
  wt2[idx] = conv_w[o * RPN_K + c * 9 + tap];
}

// ---------------------------------------------------------------------------
// 1. conv3x3 + bias + relu as implicit GEMM via V_WMMA_F32_16X16X4_F32.
//    Block = 256 thr (8 waves) computes 64(M) x 128(N); wave = 32 x 32
//    (2x2 grid of 16x16 f32 accumulators).  K order: tap-major, c-minor.
// ---------------------------------------------------------------------------
__global__ __launch_bounds__(256) void conv_wmma_kernel(
    const float* __restrict__ feat,    // [4,512,25,25]
    const float* __restrict__ wt2,     // [512][9*512] tap-major
    const float* __restrict__ conv_b,  // [512]
    float* __restrict__ xt)            // [4,512,625]
{
  __shared__ float As[64][RPN_KB + 2];   // pad 2: 8B-aligned rows, no conflicts
  __shared__ float Bs[128][RPN_KB + 4];  // k-minor; pad 4: 16B-aligned rows

  const int tid  = threadIdx.x;
  const int lane = tid & 31;
  const int wid  = tid >> 5;          // 0..7
  const int half = lane >> 4;         // 0/1 (ISA f32 frag half-lane split)
  const int lr   = lane & 15;

  const int waveM = (wid & 1) * 32;
  const int waveN = (wid >> 1) * 32;

  const int mBase = blockIdx.y * 64;
  const int nBase = blockIdx.x * 128;

  // A (im2col) loader mapping: thread -> (m row, 8 consecutive c)
  const int mA  = tid & 63;
  const int kA0 = (tid >> 6) * 8;
  const int mg  = mBase + mA;
  const bool mval = (mg < RPN_M);
  const int bb  = mg / RPN_HW;
  const int hw  = mg - bb * RPN_HW;
  const int y0  = hw / 25;
  const int x0  = hw - y0 * 25;
  const float* fbase = feat + (size_t)bb * RPN_C * RPN_HW;

  v8f acc[2][2];
#pragma unroll
  for (int i = 0; i < 2; ++i)
#pragma unroll
    for (int j = 0; j < 2; ++j) acc[i][j] = (v8f){0.f,0.f,0.f,0.f,0.f,0.f,0.f,0.f};

  for (int tap = 0; tap < 9; ++tap) {
    const int ky = tap / 3, kx = tap - ky * 3;
    const int yy = y0 + ky - 1;
    const int xx = x0 + kx - 1;
    const bool valid = mval && (unsigned)yy < 25u && (unsigned)xx < 25u;
    const float* ap = fbase + yy * 25 + xx;     // + c*625 per channel
    const float* wp = wt2 + tap * RPN_C;        // row offset inside K

    for (int cb = 0; cb < RPN_C; cb += RPN_KB) {
      // ---- stage A tile: pure stride-625 channel reads, no index math ----
#pragma unroll
      for (int kk = 0; kk < 8; ++kk) {
        int c = cb + kA0 + kk;
        float v = 0.f;
        if (valid) v = ap[(size_t)c * RPN_HW];
        As[mA][kA0 + kk] = v;
      }
      // ---- stage B tile (k-minor), float4 in and out ----
      {
        int q = tid;
#pragma unroll
        for (int rep = 0; rep < 4; ++rep) {
          int nn  = q >> 3;             // 0..127
          int kk4 = (q & 7) * 4;        // 0..28
          const float4 w4 = *(const float4*)
              &wp[(size_t)(nBase + nn) * RPN_K + cb + kk4];
          *(float4*)&Bs[nn][kk4] = w4;
          q += 256;
        }
      }
      __syncthreads();

      // ---- MMA: 8 k-steps x (2 M-frags x 2 N-frags) per K block ----
#pragma unroll
      for (int ks = 0; ks < RPN_KB; ks += 4) {
        // A 16x4 f32: lane<16 {A[l,k0],A[l,k0+1]}, lane>=16 {A[l-16,k0+2],...}
        v2f a[2], b[2];
#pragma unroll
        for (int f = 0; f < 2; ++f)
          a[f] = *(const v2f*)&As[waveM + f * 16 + lr][ks + half * 2];
        // B 4x16 f32 (k-minor rows): lane<16 {B[k0,l],B[k0+1,l]}, ...
#pragma unroll
        for (int g = 0; g < 2; ++g)
          b[g] = *(const v2f*)&Bs[waveN + g * 16 + lr][ks + half * 2];
#pragma unroll
        for (int f = 0; f < 2; ++f)
#pragma unroll
          for (int g = 0; g < 2; ++g)
            acc[f][g] = __builtin_amdgcn_wmma_f32_16x16x4_f32(
                false, a[f], false, b[g], (short)0, acc[f][g], false, false);
      }
      __syncthreads();
    }
  }

  // ---- epilogue: bias + relu, scatter to NCHW layout xt[b][o][hw] ----
#pragma unroll
  for (int g = 0; g < 2; ++g) {
    const int n    = nBase + waveN + g * 16 + lr;
    const float bi = conv_b[n];
#pragma unroll
    for (int f = 0; f < 2; ++f) {
#pragma unroll
      for (int r = 0; r < 8; ++r) {
        // C/D layout: VGPR r, lane<16 -> M=r ; lane>=16 -> M=r+8 (N=lane%16)
        int m = mBase + waveM + f * 16 + r + half * 8;
        if (m < RPN_M) {
          int b0 = m / RPN_HW, h0 = m - b0 * RPN_HW;
          float v = acc[f][g][r] + bi;
          xt[(size_t)(b0 * RPN_C + n) * RPN_HW + h0] = v > 0.f ? v : 0.f;
        }
      }
    }
  }
}

// ---------------------------------------------------------------------------
// 2. cls/bbox heads: per spatial location, 54 accumulators in registers,
//    weights fetched with uniform (scalar) addresses.
// ---------------------------------------------------------------------------
__global__ __launch_bounds__(256) void heads_kernel(
    const float* __restrict__ xt,
    const float* __restrict__ cls_w, const float* __restrict__ cls_b,
    const float* __restrict__ bbox_w, const float* __restrict__ bbox_b,
    float* __restrict__ obj,   // [b][hw][18]
    float* __restrict__ dlt)   // [b][hw][36]
{
  int m = blockIdx.x * blockDim.x + threadIdx.x;
  if (m >= RPN_M) return;
  int b = m / RPN_HW, hw = m - b * RPN_HW;
  float acc[54];
#pragma unroll
  for (int o = 0; o < 54; ++o) acc[o] = 0.f;
  const float* xp = xt + (size_t)(b * RPN_C) * RPN_HW + hw;
  for (int c = 0; c < RPN_C; ++c) {
    float xv = xp[(size_t)c * RPN_HW];
#pragma unroll
    for (int o = 0; o < 18; ++o) acc[o] += xv * cls_w[o * RPN_C + c];
#pragma unroll
    for (int o = 0; o < 36; ++o) acc[18 + o] += xv * bbox_w[o * RPN_C + c];
  }
#pragma unroll
  for (int o = 0; o < 18; ++o)
    obj[(size_t)(b * RPN_HW + hw) * 18 + o] = acc[o] + cls_b[o];
#pragma unroll
  for (int o = 0; o < 36; ++o)
    dlt[(size_t)(b * RPN_HW + hw) * 36 + o] = acc[18 + o] + bbox_b[o];
}

// ---------------------------------------------------------------------------
// 3. decode: anchors on the fly, apply deltas, clip, softmax[...,1] score.
// ---------------------------------------------------------------------------
__global__ void decode_kernel(const float* __restrict__ obj,
                              const float* __restrict__ dlt,
                              const int* __restrict__ img_h_p,
                              const int* __restrict__ img_w_p,
                              float* __restrict__ props,   // [b][n][4]
                              float* __restrict__ scores)  // [b][n]
{
  int idx = blockIdx.x * blockDim.x + threadIdx.x;
  if (idx >= RPN_B * RPN_NBOX) return;
  int b  = idx / RPN_NBOX;
  int n  = idx - b * RPN_NBOX;
  int hw = n / 9, a = n - hw * 9;
  int y  = hw / 25, x = hw - y * 25;

  const float scl[3] = {64.f, 128.f, 256.f};
  const float rat[3] = {0.5f, 1.f, 2.f};
  float s  = scl[a / 3];
  float r  = rat[a % 3];
  float sq = sqrtf(r);
  float aw = s * sq, ah = s / sq;
  float cx = (float)(x * 32), cy = (float)(y * 32);

  const float* d = &dlt[(size_t)(b * RPN_HW + hw) * 36 + a * 4];
  float pcx = d[0] * aw + cx;
  float pcy = d[1] * ah + cy;
  float pw  = expf(d[2]) * aw;
  float ph  = expf(d[3]) * ah;

  float W = (float)img_w_p[0], H = (float)img_h_p[0];
  float x1 = fminf(fmaxf(pcx - 0.5f * pw, 0.f), W);
  float y1 = fminf(fmaxf(pcy - 0.5f * ph, 0.f), H);
  float x2 = fminf(fmaxf(pcx + 0.5f * pw, 0.f), W);
  float y2 = fminf(fmaxf(pcy + 0.5f * ph, 0.f), H);
  ((float4*)props)[idx] = make_float4(x1, y1, x2, y2);

  const float* o = &obj[(size_t)(b * RPN_HW + hw) * 18 + a * 2];
  scores[idx] = 1.f / (1.f + expf(o[0] - o[1]));  // == softmax(...)[1]
}

// ---------------------------------------------------------------------------
// 4. rank sort (stable, matches argsort(-scores)): rank = #(better j).
// ---------------------------------------------------------------------------
__global__ __launch_bounds__(256) void rank_kernel(
    const float* __restrict__ scores, int* __restrict__ order) {
  __shared__ float ssc[RPN_NBOX];
  int b = blockIdx.y;
  const float* sp = scores + (size_t)b * RPN_NBOX;
  for (int i = threadIdx.x; i < RPN_NBOX; i += 256) ssc[i] = sp[i];
  __syncthreads();
  int i = blockIdx.x * 256 + threadIdx.x;
  if (i >= RPN_NBOX) return;
  float si = ssc[i];
  int rank = 0;
  for (int j = 0; j < RPN_NBOX; ++j) {
    float sj = ssc[j];
    rank += (sj > si) || (sj == si && j < i);
  }
  order[(size_t)b * RPN_NBOX + rank] = i;
}

// ---------------------------------------------------------------------------
// 5. gather boxes into sorted order.
// ---------------------------------------------------------------------------
__global__ void gather_kernel(const float* __restrict__ props,
                              const int* __restrict__ order,
                              float* __restrict__ sboxes) {
  int idx = blockIdx.x * blockDim.x + threadIdx.x;
  if (idx >= RPN_B * RPN_NBOX) return;
  int b = idx / RPN_NBOX;
  int src = order[idx] + b * RPN_NBOX;
  ((float4*)sboxes)[idx] = ((const float4*)props)[src];
}

// ---------------------------------------------------------------------------
// 6. suppression bitmask: mask[b][i][w] bit set iff iou(sb[i], sb[j]) > thr,
//    j = w*32+bit, j > i   (torchvision semantics).
// ---------------------------------------------------------------------------
__global__ void mask_kernel(const float* __restrict__ sboxes,
                            unsigned* __restrict__ mask) {
  int idx = blockIdx.x * blockDim.x + threadIdx.x;
  if (idx >= RPN_B * RPN_NBOX * RPN_MW) return;
  int w = idx % RPN_MW;
  int t = idx / RPN_MW;
  int i = t % RPN_NBOX;
  int b = t / RPN_NBOX;

  const float4 bi = ((const float4*)sboxes)[b * RPN_NBOX + i];
  float ai = (bi.z - bi.x) * (bi.w - bi.y);
  unsigned bits = 0u;
  int j0 = w * 32;
#pragma unroll 4
  for (int bit = 0; bit < 32; ++bit) {
    int j = j0 + bit;
    if (j > i && j < RPN_NBOX) {
      const float4 bj = ((const float4*)sboxes)[b * RPN_NBOX + j];
      float aj  = (bj.z - bj.x) * (bj.w - bj.y);
      float ix1 = fmaxf(bi.x, bj.x), iy1 = fmaxf(bi.y, bj.y);
      float ix2 = fminf(bi.z, bj.z), iy2 = fminf(bi.w, bj.w);
      float iw  = fmaxf(ix2 - ix1, 0.f);
      float ih  = fmaxf(iy2 - iy1, 0.f);
      float inter = iw * ih;
      float iou = inter / (ai + aj - inter);
      if (iou > NMS_THR) bits |= (1u << bit);
    }
  }
  mask[idx] = bits;
}

// ---------------------------------------------------------------------------
// 7. sequential greedy scan, one wave32 per batch (LDS ops in-order per wave).
// ---------------------------------------------------------------------------
__global__ __launch_bounds__(32) void nms_scan_kernel(
    const unsigned* __restrict__ mask, const int* __restrict__ order,
    float* __restrict__ keep) {
  __shared__ unsigned removed[RPN_MW];
  int b = blockIdx.x;
  int t = threadIdx.x;
  for (int w = t; w < RPN_MW; w += 32) removed[w] = 0u;
  __syncthreads();
  for (int i = 0; i < RPN_NBOX; ++i) {
    unsigned rw = removed[i >> 5];                 // uniform across lanes
    bool alive = ((rw >> (i & 31)) & 1u) == 0u;
    __syncthreads();
    if (alive) {
      const unsigned* row = mask + (size_t)(b * RPN_NBOX + i) * RPN_MW;
      for (int w = t; w < RPN_MW; w += 32) removed[w] |= row[w];
    }
    __syncthreads();
  }
  for (int i = t; i < RPN_NBOX; i += 32) {
    unsigned rw = removed[i >> 5];
    float k = ((rw >> (i & 31)) & 1u) ? 0.f : 1.f;
    keep[(size_t)b * RPN_NBOX + order[b * RPN_NBOX + i]] = k;
  }
}

// ---------------------------------------------------------------------------
// 8. finalize: out = props * keep (original box order).
// ---------------------------------------------------------------------------
__global__ void finalize_kernel(const float* __restrict__ props,
                                const float* __restrict__ keep,
                                float* __restrict__ out) {
  int idx = blockIdx.x * blockDim.x + threadIdx.x;
  if (idx >= RPN_B * RPN_NBOX) return;
  float k = keep[idx];
  float4 p = ((const float4*)props)[idx];
  ((float4*)out)[idx] = make_float4(p.x * k, p.y * k, p.z * k, p.w * k);
}

// ---------------------------------------------------------------------------
extern "C" void kernel_launch(void* const* d_in, const int* in_sizes, int n_in,
                              void* d_out, int out_size, void* d_ws,
                              size_t ws_size, hipStream_t stream) {
  (void)in_sizes; (void)n_in; (void)out_size; (void)ws_size;
  const float* feat   = (const float*)d_in[0];
  const float* conv_w = (const float*)d_in[1];
  const float* conv_b = (const float*)d_in[2];
  const float* cls_w  = (const float*)d_in[3];
  const float* cls_b  = (const float*)d_in[4];
  const float* bbox_w = (const float*)d_in[5];
  const float* bbox_b = (const float*)d_in[6];
  const int*   img_h  = (const int*)d_in[7];
  const int*   img_w  = (const int*)d_in[8];

  char* ws = (char*)d_ws;
  size_t off = 0;
  auto carve = [&](size_t bytes) {
    char* p = ws + off;
    off = (off + bytes + 255) & ~(size_t)255;
    return p;
  };
  float*    xt     = (float*)carve((size_t)RPN_B * RPN_C * RPN_HW * 4);      // 5.12 MB
  float*    wt2    = (float*)carve((size_t)RPN_N * RPN_K * 4);               // 9.44 MB
  float*    obj    = (float*)carve((size_t)RPN_B * RPN_HW * 18 * 4);
  float*    dlt    = (float*)carve((size_t)RPN_B * RPN_HW * 36 * 4);
  float*    props  = (float*)carve((size_t)RPN_B * RPN_NBOX * 4 * 4);
  float*    scores = (float*)carve((size_t)RPN_B * RPN_NBOX * 4);
  int*      order  = (int*)carve((size_t)RPN_B * RPN_NBOX * 4);
  float*    sboxes = (float*)carve((size_t)RPN_B * RPN_NBOX * 4 * 4);
  unsigned* maskb  = (unsigned*)carve((size_t)RPN_B * RPN_NBOX * RPN_MW * 4); // 15.8 MB
  float*    keep   = (float*)carve((size_t)RPN_B * RPN_NBOX * 4);

  // 0. weight repack [O][c*9+tap] -> [O][tap*512+c]
  wtrans_kernel<<<(RPN_N * RPN_K + 255) / 256, 256, 0, stream>>>(conv_w, wt2);
  // 1. conv3x3 + relu via f32 WMMA (64x128 block tile, 32x32 per wave)
  conv_wmma_kernel<<<dim3(RPN_N / 128, (RPN_M + 63) / 64), 256, 0, stream>>>(
      feat, wt2, conv_b, xt);
  // 2. heads
  heads_kernel<<<(RPN_M + 255) / 256, 256, 0, stream>>>(
      xt, cls_w, cls_b, bbox_w, bbox_b, obj, dlt);
  // 3. decode + scores
  decode_kernel<<<(RPN_B * RPN_NBOX + 255) / 256, 256, 0, stream>>>(
      obj, dlt, img_h, img_w, props, scores);
  // 4. stable descending rank sort
  rank_kernel<<<dim3((RPN_NBOX + 255) / 256, RPN_B), 256, 0, stream>>>(
      scores, order);
  // 5. gather sorted boxes
  gather_kernel<<<(RPN_B * RPN_NBOX + 255) / 256, 256, 0, stream>>>(
      props, order, sboxes);
  // 6. pairwise IoU suppression bitmask
  mask_kernel<<<(RPN_B * RPN_NBOX * RPN_MW + 255) / 256, 256, 0, stream>>>(
      sboxes, maskb);
  // 7. sequential greedy scan (1 wave / batch)
  nms_scan_kernel<<<RPN_B, 32, 0, stream>>>(maskb, order, keep);
  // 8. masked output
  finalize_kernel<<<(RPN_B * RPN_NBOX + 255) / 256, 256, 0, stream>>>(
      props, keep, (float*)d_out);
}